// ClusterPool_18708877542040
// MI455X (gfx1250) — compile-verified
//
#include <hip/hip_runtime.h>
#include <math.h>

// ---------------- problem constants ----------------
#define D_FEAT       128
#define TILE_ROWS    40          // rows per TDM tile
#define NBUF         3           // triple-buffered LDS ring: 3*40*128*4 = 60 KB
#define BLOCK        256         // 8 wave32 per workgroup
#define WAVES        (BLOCK / 32)
#define ROWS_PER_WAVE (TILE_ROWS / WAVES)   // 5

typedef unsigned int u32;
typedef unsigned long long u64;
typedef __attribute__((ext_vector_type(4))) u32 v4u;
typedef __attribute__((ext_vector_type(8))) int v8i;
typedef __attribute__((ext_vector_type(4))) int v4i;

// ---- Tensor Data Mover: DMA a [nrows x 128] fp32 tile (row-major, stride 128)
// ---- from global memory into LDS at byte offset lds_off.
// ---- amdgpu-toolchain (clang-23) 6-arg builtin:
// ----   (uint32x4 g0, int32x8 g1, int32x4 g2, int32x4 g3, int32x8, i32 cpol)
__device__ __forceinline__ void tdm_load_tile(u32 lds_off, const float* gsrc, u32 nrows) {
    u64 ga = (u64)(uintptr_t)gsrc;
    // D# group 0: count=1 | lds_addr | global_addr[56:0] | type=2
    v4u g0;
    g0[0] = 1u;                                       // count=1, user mode, no gather
    g0[1] = lds_off;                                  // LDS byte address
    g0[2] = (u32)(ga & 0xFFFFFFFFu);                  // global_addr[31:0]
    g0[3] = (u32)((ga >> 32) & 0x1FFFFFFu) | (2u << 30); // global_addr[56:32] | type=2
    // D# group 1
    const u32 dim0 = D_FEAT;        // tensor_dim0 = 128 elements per row
    const u32 str0 = D_FEAT;        // tensor_dim0_stride = 128 elements
    v8i g1;
    g1[0] = (int)(2u << 16);                                  // mask=0, data_size=2 (4B)
    g1[1] = (int)((dim0 & 0xFFFFu) << 16);                    // tensor_dim0[15:0]
    g1[2] = (int)((dim0 >> 16) | ((nrows & 0xFFFFu) << 16));  // dim0[31:16] | tensor_dim1[15:0]
    g1[3] = (int)((nrows >> 16) | ((D_FEAT & 0xFFFFu) << 16));// dim1[31:16] | tile_dim0=128
    g1[4] = (int)(nrows & 0xFFFFu);                           // tile_dim1=nrows, tile_dim2=0
    g1[5] = (int)str0;                                        // tensor_dim0_stride[31:0]
    g1[6] = 0;                                                // stride hi / dim1_stride lo
    g1[7] = 0;                                                // dim1_stride hi (unused, 2D)
    v4i z4 = {0, 0, 0, 0};                                    // groups 2/3: unused (2D)
    v8i z8 = {0, 0, 0, 0, 0, 0, 0, 0};
    __builtin_amdgcn_tensor_load_to_lds(g0, g1, z4, z4, z8, /*cpol=*/0);
}

// ---- native no-return f32 max atomic at device scope (merges partial segment maxima)
__device__ __forceinline__ void atomic_max_f32(float* p, float v) {
    asm volatile("global_atomic_max_num_f32 %0, %1, off scope:SCOPE_DEV"
                 :: "v"(p), "v"(v) : "memory");
}

__device__ __forceinline__ void flush_seg(float* __restrict__ out, int cseg, int lane,
                                          float4 m) {
    if (cseg < 0) return;
    float* p = out + (size_t)cseg * D_FEAT + lane * 4;
    atomic_max_f32(p + 0, m.x);
    atomic_max_f32(p + 1, m.y);
    atomic_max_f32(p + 2, m.z);
    atomic_max_f32(p + 3, m.w);
}

__global__ void ClusterPool_init_kernel(float* __restrict__ out, int n) {
    int i = blockIdx.x * blockDim.x + threadIdx.x;
    if (i < n) out[i] = -__builtin_inff();
}

__global__ void __launch_bounds__(BLOCK)
ClusterPool_segmax_kernel(const float* __restrict__ feat,
                          const int* __restrict__ seg,
                          float* __restrict__ out,
                          int n_points, int rows_per_wg) {
    __shared__ float lds[NBUF * TILE_ROWS * D_FEAT];

    const long long row0 = (long long)blockIdx.x * rows_per_wg;
    if (row0 >= n_points) return;
    const int rows   = min(rows_per_wg, (int)(n_points - row0));
    const int ntiles = (rows + TILE_ROWS - 1) / TILE_ROWS;

    const int tid  = threadIdx.x;
    const int wave = tid >> 5;
    const int lane = tid & 31;
    const u32 lds_base = (u32)(uintptr_t)&lds[0];

    // Prologue: get two tiles in flight (TENSORcnt lives on wave 0 only).
    if (wave == 0) {
        for (int t = 0; t < 2 && t < ntiles; ++t) {
            const int tr0 = t * TILE_ROWS;
            const int nr  = min(TILE_ROWS, rows - tr0);
            tdm_load_tile(lds_base + (u32)(t % NBUF) * (TILE_ROWS * D_FEAT * 4u),
                          feat + (row0 + tr0) * (long long)D_FEAT, (u32)nr);
        }
    }

    float4 cmax = make_float4(-__builtin_inff(), -__builtin_inff(),
                              -__builtin_inff(), -__builtin_inff());
    int cseg = -1;

    for (int t = 0; t < ntiles; ++t) {
        if (wave == 0) {
            if (t + 1 < ntiles) __builtin_amdgcn_s_wait_tensorcnt(1); // tile t done
            else                __builtin_amdgcn_s_wait_tensorcnt(0); // drain
        }
        __syncthreads();   // publish tile t's LDS data to all waves

        const int tr0 = t * TILE_ROWS;
        const int nr  = min(TILE_ROWS, rows - tr0);
        const float4* buf = (const float4*)&lds[(t % NBUF) * TILE_ROWS * D_FEAT];

        const int rbeg = wave * ROWS_PER_WAVE;
        const int rend = min(rbeg + ROWS_PER_WAVE, nr);
        for (int r = rbeg; r < rend; ++r) {
            const long long grow = row0 + tr0 + r;
            const int s = seg[grow];                     // wave-uniform broadcast load
            const float4 v = buf[r * (D_FEAT / 4) + lane]; // ds_load_b128
            if (s != cseg) {
                flush_seg(out, cseg, lane, cmax);
                cseg = s;
                cmax = v;
            } else {
                cmax.x = fmaxf(cmax.x, v.x);
                cmax.y = fmaxf(cmax.y, v.y);
                cmax.z = fmaxf(cmax.z, v.z);
                cmax.w = fmaxf(cmax.w, v.w);
            }
        }
        __syncthreads();   // everyone done reading buf[t % NBUF] and earlier

        if (wave == 0) {
            const int tt = t + 2;                        // refill ring slot (t+2)%3
            if (tt < ntiles) {
                const int tr = tt * TILE_ROWS;
                const int nr2 = min(TILE_ROWS, rows - tr);
                tdm_load_tile(lds_base + (u32)(tt % NBUF) * (TILE_ROWS * D_FEAT * 4u),
                              feat + (row0 + tr) * (long long)D_FEAT, (u32)nr2);
            }
        }
    }

    flush_seg(out, cseg, lane, cmax);  // final partial segment
}

extern "C" void kernel_launch(void* const* d_in, const int* in_sizes, int n_in,
                              void* d_out, int out_size, void* d_ws, size_t ws_size,
                              hipStream_t stream) {
    const float* feat = (const float*)d_in[0];
    const int*   seg  = (const int*)d_in[1];
    float*       out  = (float*)d_out;

    const int n_points = in_sizes[0] / D_FEAT;

    // 1) output = -inf (max identity; empty clusters stay -inf like jax.segment_max)
    const int init_blocks = (out_size + BLOCK - 1) / BLOCK;
    ClusterPool_init_kernel<<<init_blocks, BLOCK, 0, stream>>>(out, out_size);

    // 2) streaming segment-max, contiguous row chunks per workgroup
    int num_wg = 2048;
    if (num_wg > n_points) num_wg = n_points > 0 ? n_points : 1;
    const int rows_per_wg = (n_points + num_wg - 1) / num_wg;
    ClusterPool_segmax_kernel<<<num_wg, BLOCK, 0, stream>>>(feat, seg, out,
                                                            n_points, rows_per_wg);
}